// PartialConnection_81277961109693
// MI455X (gfx1250) — compile-verified
//
#include <hip/hip_runtime.h>
#include <hip/hip_bf16.h>
#include <cstdint>
#include <cstddef>

#define UNITS     17824
#define DEG       8
#define INPUT_DIM 17824
#define NEDGE     (UNITS * DEG)      // 142592
#define B_TOTAL   512

#define NT      256                  // nodes per block (== blockDim.x)
#define TILE_E  (NT * DEG)           // 2048 edges staged per block
#define BB      32                   // batch rows per block
#define NTILES  ((UNITS + NT - 1) / NT)   // 70

__global__ __launch_bounds__(256) void pc_gather_segsum_kernel(
    const float* __restrict__ x,     // [B, INPUT_DIM]
    const float* __restrict__ kern,  // [E]
    const float* __restrict__ bias,  // [E]
    const int*   __restrict__ col,   // [E]
    float* __restrict__ out)         // [B, UNITS]
{
    __shared__ __align__(16) float s_k[TILE_E];
    __shared__ __align__(16) float s_b[TILE_E];
    __shared__ __align__(16) int   s_c[TILE_E];

    const int tile   = blockIdx.x;
    const int tileE0 = tile * TILE_E;
    const uint32_t maxByte = (uint32_t)NEDGE * 4u - 16u;   // last valid 16B chunk

    // ---- Phase 1: stage this tile's per-edge constants into LDS via the
    // CDNA5 async global->LDS DMA path (tracked by ASYNCcnt). 24 KB total:
    // 512 chunks of 16B per array, 256 threads -> 2 chunks/thread/array.
    #pragma unroll
    for (int r = 0; r < 2; ++r) {
        const int ch = (int)threadIdx.x + r * 256;          // chunk id 0..511
        uint32_t go = (uint32_t)tileE0 * 4u + (uint32_t)ch * 16u;
        if (go > maxByte) go = maxByte;                     // clamp for tail tile
        const uint32_t lk = (uint32_t)(uintptr_t)&s_k[ch * 4];
        const uint32_t lb = (uint32_t)(uintptr_t)&s_b[ch * 4];
        const uint32_t lc = (uint32_t)(uintptr_t)&s_c[ch * 4];
        asm volatile("global_load_async_to_lds_b128 %0, %1, %2 offset:0"
                     :: "v"(lk), "v"(go), "s"(kern) : "memory");
        asm volatile("global_load_async_to_lds_b128 %0, %1, %2 offset:0"
                     :: "v"(lb), "v"(go), "s"(bias) : "memory");
        asm volatile("global_load_async_to_lds_b128 %0, %1, %2 offset:0"
                     :: "v"(lc), "v"(go), "s"(col)  : "memory");
    }
    asm volatile("s_wait_asynccnt 0" ::: "memory");
    __syncthreads();

    // ---- Phase 2: one node per lane. Consecutive lanes -> consecutive nodes,
    // and col[n*8+d] = (n + d*997) % D, so per-d gathers are dense/coalesced.
    const int  node  = tile * NT + (int)threadIdx.x;
    const bool valid = (node < UNITS);

    float k[DEG];
    int   c[DEG];
    float bsum = 0.0f;
    if (valid) {
        const int e0 = (int)threadIdx.x * DEG;
        #pragma unroll
        for (int d = 0; d < DEG; ++d) {
            k[d] = s_k[e0 + d];
            c[d] = s_c[e0 + d];
            bsum += s_b[e0 + d];
        }
    } else {
        #pragma unroll
        for (int d = 0; d < DEG; ++d) { k[d] = 0.0f; c[d] = 0; }
    }

    const int b0 = (int)blockIdx.y * BB;
    #pragma unroll 2
    for (int bi = 0; bi < BB; ++bi) {
        const int b = b0 + bi;
        const float* __restrict__ xr = x + (size_t)b * INPUT_DIM;
        float acc = bsum;
        #pragma unroll
        for (int d = 0; d < DEG; ++d)
            acc = fmaf(xr[c[d]], k[d], acc);
        if (valid)
            out[(size_t)b * UNITS + node] = fmaxf(acc, 0.0f);
    }
}

extern "C" void kernel_launch(void* const* d_in, const int* in_sizes, int n_in,
                              void* d_out, int out_size, void* d_ws, size_t ws_size,
                              hipStream_t stream) {
    (void)in_sizes; (void)n_in; (void)out_size; (void)d_ws; (void)ws_size;
    const float* x    = (const float*)d_in[0];
    const float* kern = (const float*)d_in[1];
    const float* bias = (const float*)d_in[2];
    const int*   col  = (const int*)d_in[3];
    // d_in[4] (seg_idx) is repeat(arange(UNITS), 8): edges of node n are the
    // contiguous slice [8n, 8n+8), handled implicitly by the tiling.
    float* out = (float*)d_out;

    dim3 grid(NTILES, B_TOTAL / BB);   // 70 x 16 blocks
    dim3 block(NT);                    // 256 threads = 8 wave32
    pc_gather_segsum_kernel<<<grid, block, 0, stream>>>(x, kern, bias, col, out);
}